// OCM_23416161698500
// MI455X (gfx1250) — compile-verified
//
#include <hip/hip_runtime.h>
#include <hip/hip_bf16.h>

typedef float v2f __attribute__((ext_vector_type(2)));
typedef float v8f __attribute__((ext_vector_type(8)));
typedef int   v4i __attribute__((vector_size(16)));

#define BATCH 8
#define NDIM  32768
#define CATD  128
#define FEATD 128
#define NKEYS 64
#define NTILE 128
#define LDS_STRIDE 132   // 128 + 4 pad -> conflict-free ds b64/2addr B-fragments

#if defined(__gfx1250__) && __has_builtin(__builtin_amdgcn_global_load_async_to_lds_b128)
#define USE_ASYNC_LDS 1
#else
#define USE_ASYNC_LDS 0
#endif

typedef __attribute__((address_space(1))) v4i* gv4i_p;   // global (AS1)
typedef __attribute__((address_space(3))) v4i* lv4i_p;   // LDS (AS3)

// ---------------------------------------------------------------------------
// Kernel 1: out[b,f,n] = sum_c input[b,n,c] * weight[c,f]
//   D(16x16) = A(16x4: W^T tile) x B(4x16: X^T tile), K swept 128/4 = 32 steps.
//   Block: 256 thr = 8 waves; wave w owns FEAT rows [16w,16w+16), block owns a
//   128-wide n tile of one batch. X tile staged to LDS via async direct
//   global->LDS (ASYNCcnt) when available; W fragments read through L2.
// ---------------------------------------------------------------------------
__global__ __launch_bounds__(256) void gemm_bfn_wmma_kernel(
    const float* __restrict__ input,
    const float* __restrict__ weight,
    float* __restrict__ out) {
  extern __shared__ float smem[];  // NTILE * LDS_STRIDE floats

  const int tid  = threadIdx.x;
  const int lane = tid & 31;
  const int wave = tid >> 5;       // 0..7 -> FEAT strip
  const int hi   = lane >> 4;      // 0/1: which K-pair this lane holds
  const int fl   = lane & 15;      // M (A) / N (B) index within tile

  const int b  = blockIdx.x >> 8;            // 256 n-tiles per batch
  const int n0 = (blockIdx.x & 255) * NTILE;

  // ---- stage X tile (NTILE rows x CATD cols fp32) into LDS ----
  const float* src = input + ((size_t)b * NDIM + n0) * CATD;
#if USE_ASYNC_LDS
  #pragma unroll
  for (int p = 0; p < 16; ++p) {
    int idx = p * 256 + tid;       // 0..4095 16-byte chunks
    int r   = idx >> 5;            // tile row (n)
    int j   = idx & 31;            // float4 within row (c/4)
    __builtin_amdgcn_global_load_async_to_lds_b128(
        (gv4i_p)(src + (size_t)r * CATD + 4 * j),
        (lv4i_p)&smem[r * LDS_STRIDE + 4 * j],
        0, 0);
  }
#if __has_builtin(__builtin_amdgcn_s_wait_asynccnt)
  __builtin_amdgcn_s_wait_asynccnt(0);
#else
  asm volatile("s_wait_asynccnt 0x0" ::: "memory");
#endif
  __syncthreads();
#else
  const float4* src4 = (const float4*)src;
  #pragma unroll
  for (int p = 0; p < 16; ++p) {
    int idx = p * 256 + tid;       // 0..4095 float4 slots
    int r   = idx >> 5;            // tile row (n)
    int j   = idx & 31;            // float4 within row (c/4)
    float4 v = src4[r * 32 + j];
    *(float4*)&smem[r * LDS_STRIDE + j * 4] = v;
  }
  __syncthreads();
#endif

  const int f = wave * 16 + fl;    // this lane's FEAT column in W

  v8f acc[8];
  #pragma unroll
  for (int s = 0; s < 8; ++s) acc[s] = (v8f)(0.0f);

  #pragma unroll
  for (int k = 0; k < 32; ++k) {
    const int c = 4 * k + 2 * hi;
    // A fragment: A(M=f, K=c..c+1) = weight[c][f], weight[c+1][f]
    v2f a;
    a.x = weight[(size_t)c * FEATD + f];
    a.y = weight[(size_t)(c + 1) * FEATD + f];
    #pragma unroll
    for (int s = 0; s < 8; ++s) {
      // B fragment: B(K=c..c+1, N=n) = X[n][c], X[n][c+1] from LDS
      const v2f bfr = *(const v2f*)&smem[(16 * s + fl) * LDS_STRIDE + c];
      acc[s] = __builtin_amdgcn_wmma_f32_16x16x4_f32(
          false, a, false, bfr, (short)0, acc[s], false, false);
    }
  }

  // ---- store D: out[b, f, n]; lanes carry 16 consecutive n -> coalesced ----
  float* outb = out + (size_t)b * FEATD * NDIM;
  #pragma unroll
  for (int s = 0; s < 8; ++s) {
    const int n = n0 + 16 * s + fl;
    #pragma unroll
    for (int r = 0; r < 8; ++r) {
      const int ff = wave * 16 + 8 * hi + r;   // D VGPR r -> M = r (+8 hi half)
      outb[(size_t)ff * NDIM + n] = acc[s][r];
    }
  }
}

// ---------------------------------------------------------------------------
// Kernel 2: new_weight = weight; for each key k:
//   row = 0.5*weight[key] + 0.5*values[k]; new_weight[key][:] = ||row||_2
// Single block, 8 waves; wave handles 8 keys; wave32 shfl_xor reduction.
// ---------------------------------------------------------------------------
__global__ __launch_bounds__(256) void ema_norm_kernel(
    const float* __restrict__ weight,
    const int* __restrict__ keys,
    const float* __restrict__ values,
    float* __restrict__ new_weight) {
  const int tid = threadIdx.x;
  for (int i = tid; i < CATD * FEATD; i += 256)
    new_weight[i] = weight[i];
  __syncthreads();

  const int wave = tid >> 5, lane = tid & 31;
  for (int kk = wave; kk < NKEYS; kk += 8) {
    const int key = keys[kk];
    float ss = 0.0f;
    #pragma unroll
    for (int j = 0; j < 4; ++j) {
      int e = lane + 32 * j;
      float nv = 0.5f * weight[(size_t)key * FEATD + e] +
                 0.5f * values[(size_t)kk * FEATD + e];
      ss += nv * nv;
    }
    #pragma unroll
    for (int off = 16; off > 0; off >>= 1)
      ss += __shfl_xor(ss, off, 32);
    const float nrm = sqrtf(ss);
    #pragma unroll
    for (int j = 0; j < 4; ++j)
      new_weight[(size_t)key * FEATD + lane + 32 * j] = nrm;
  }
}

extern "C" void kernel_launch(void* const* d_in, const int* in_sizes, int n_in,
                              void* d_out, int out_size, void* d_ws, size_t ws_size,
                              hipStream_t stream) {
  const float* input  = (const float*)d_in[0];  // (B, N, CAT) fp32
  const float* weight = (const float*)d_in[1];  // (CAT, FEAT) fp32
  const int*   keys   = (const int*)d_in[2];    // (K,) int32
  const float* values = (const float*)d_in[3];  // (K, FEAT) fp32

  float* out        = (float*)d_out;                          // (B, FEAT, N)
  float* new_weight = out + (size_t)BATCH * FEATD * NDIM;     // (CAT, FEAT)

  dim3 grid(BATCH * (NDIM / NTILE));   // 2048 blocks
  size_t shmem = (size_t)NTILE * LDS_STRIDE * sizeof(float);  // 67.5 KB

  hipLaunchKernelGGL(gemm_bfn_wmma_kernel, grid, dim3(256), shmem, stream,
                     input, weight, out);
  hipLaunchKernelGGL(ema_norm_kernel, dim3(1), dim3(256), 0, stream,
                     weight, keys, values, new_weight);
}